// ModelNew_3556232922273
// MI455X (gfx1250) — compile-verified
//
#include <hip/hip_runtime.h>
#include <math.h>
#include <stdint.h>

typedef float v2f __attribute__((ext_vector_type(2)));
typedef float v8f __attribute__((ext_vector_type(8)));

#define BATCH_N   65536
#define KDIM      2048
#define GROUP     256     // rows per max-group (BLOCK in reference)
#define THREADS   512     // 16 wave32s
#define NWAVES    16
#define KC        32      // k-chunk streamed per stage
#define RSTRIDE   36      // padded LDS row stride (dwords): conflict-free, 16B-aligned rows
#define TILE_DW   (16 * RSTRIDE)

// Issue one chunk (16 rows x 32 floats = 4 x b128/lane) global -> LDS, async.
__device__ __forceinline__ void issue_async_chunk(const float* __restrict__ x,
                                                  float* buf, int rowBase,
                                                  int kb, int lane) {
#pragma unroll
  for (int i = 0; i < 4; ++i) {
    int unit = i * 32 + lane;                 // 0..127
    int row  = unit >> 3;                     // 0..15
    int seg  = unit & 7;                      // float4 slot within row
    uint32_t ldsAddr = (uint32_t)(uintptr_t)(buf + row * RSTRIDE + seg * 4);
    uint64_t gAddr   = (uint64_t)(uintptr_t)(x + (size_t)(rowBase + row) * KDIM
                                               + kb + seg * 4);
    asm volatile("global_load_async_to_lds_b128 %0, %1, off"
                 :: "v"(ldsAddr), "v"(gAddr) : "memory");
  }
}

// Consume one chunk: 8 WMMA steps (K=4 each), two accumulator chains.
// B = weight chunk broadcast to all 16 columns -> every D element is a row dot.
__device__ __forceinline__ void consume_chunk(const float* cur, const float* wl,
                                              int kb, int l16, int half,
                                              v8f& acc0, v8f& acc1) {
#pragma unroll
  for (int ks = 0; ks < KC; ks += 8) {
    v2f a0 = *(const v2f*)(cur + l16 * RSTRIDE + ks + 2 * half);
    v2f b0 = *(const v2f*)(wl + kb + ks + 2 * half);
    acc0 = __builtin_amdgcn_wmma_f32_16x16x4_f32(false, a0, false, b0,
                                                 (short)0, acc0, false, false);
    v2f a1 = *(const v2f*)(cur + l16 * RSTRIDE + ks + 4 + 2 * half);
    v2f b1 = *(const v2f*)(wl + kb + ks + 4 + 2 * half);
    acc1 = __builtin_amdgcn_wmma_f32_16x16x4_f32(false, a1, false, b1,
                                                 (short)0, acc1, false, false);
  }
}

__global__ __launch_bounds__(THREADS)
void fused_gemv_gelu_blockmax(const float* __restrict__ x,
                              const float* __restrict__ w,
                              const float* __restrict__ bias,
                              float* __restrict__ out) {
  __shared__ float wLds[KDIM];                        // 8 KB shared weight vector
  __shared__ float xLds[NWAVES * 2 * TILE_DW];        // double-buffered per-wave tiles
  __shared__ float redBuf[NWAVES];

  const int tid  = threadIdx.x;
  const int lane = tid & 31;
  const int wave = tid >> 5;
  const int bid  = blockIdx.x;

  // Stage weight into LDS: 512 threads * float4 = 2048 floats, fully coalesced.
  ((float4*)wLds)[tid] = ((const float4*)w)[tid];

  // Zero this group's outputs; only out[bid*GROUP] receives the max (thread 0).
  if (tid > 0 && tid < GROUP) out[(size_t)bid * GROUP + tid] = 0.0f;

  __syncthreads();

  const int rowBase = bid * GROUP + wave * 16;        // this wave owns 16 batch rows
  float* buf0 = xLds + wave * (2 * TILE_DW);          // wave-private double buffer
  float* buf1 = buf0 + TILE_DW;

  const int half = lane >> 4;                         // K-pair select in a 4-wide step
  const int l16  = lane & 15;                         // A-fragment row

  v8f acc0 = {0.f,0.f,0.f,0.f,0.f,0.f,0.f,0.f};
  v8f acc1 = acc0;

  // Prologue: start streaming chunk 0.
  issue_async_chunk(x, buf0, rowBase, 0, lane);

  // Steady state: branch-free, explicitly 2x-unrolled software pipeline.
  int kb = 0;
  for (; kb < KDIM - 2 * KC; kb += 2 * KC) {
    asm volatile("s_wait_dscnt 0x0" ::: "memory");    // WAR: reads of buf1 done
    issue_async_chunk(x, buf1, rowBase, kb + KC, lane);
    asm volatile("s_wait_asynccnt 0x4" ::: "memory"); // buf0 chunk has landed
    consume_chunk(buf0, wLds, kb, l16, half, acc0, acc1);

    asm volatile("s_wait_dscnt 0x0" ::: "memory");    // WAR: reads of buf0 done
    issue_async_chunk(x, buf0, rowBase, kb + 2 * KC, lane);
    asm volatile("s_wait_asynccnt 0x4" ::: "memory"); // buf1 chunk has landed
    consume_chunk(buf1, wLds, kb + KC, l16, half, acc0, acc1);
  }

  // Epilogue: last two chunks (kb == KDIM - 2*KC), drain the pipeline.
  asm volatile("s_wait_dscnt 0x0" ::: "memory");
  issue_async_chunk(x, buf1, rowBase, kb + KC, lane);
  asm volatile("s_wait_asynccnt 0x4" ::: "memory");
  consume_chunk(buf0, wLds, kb, l16, half, acc0, acc1);
  asm volatile("s_wait_asynccnt 0x0" ::: "memory");
  consume_chunk(buf1, wLds, kb + KC, l16, half, acc0, acc1);

  // Each accumulator element is a replicated per-row dot; max over all
  // (vgpr, lane) positions == max over this wave's 16 rows (layout-agnostic).
  const float bv = bias[0];
  float m = -__builtin_inff();
#pragma unroll
  for (int j = 0; j < 8; ++j) {
    float y = (acc0[j] + acc1[j]) + bv;
    float p = y * 0.25f;                                  // avg-pool divide
    float inner = 0.7978845608f * (p + 0.044715f * p * p * p);
    float s = p * (1.0f + tanhf(inner));                  // 0.5 * scale(=2) folded
    m = fmaxf(m, s);
  }
#pragma unroll
  for (int off = 16; off > 0; off >>= 1)
    m = fmaxf(m, __shfl_xor(m, off, 32));
  if (lane == 0) redBuf[wave] = m;
  __syncthreads();
  if (tid == 0) {
    float g = redBuf[0];
#pragma unroll
    for (int i = 1; i < NWAVES; ++i) g = fmaxf(g, redBuf[i]);
    out[(size_t)bid * GROUP] = g;
  }
}

extern "C" void kernel_launch(void* const* d_in, const int* in_sizes, int n_in,
                              void* d_out, int out_size, void* d_ws, size_t ws_size,
                              hipStream_t stream) {
  const float* x    = (const float*)d_in[0];
  const float* w    = (const float*)d_in[1];
  const float* bias = (const float*)d_in[2];
  float* out        = (float*)d_out;
  (void)in_sizes; (void)n_in; (void)out_size; (void)d_ws; (void)ws_size;
  hipLaunchKernelGGL(fused_gemv_gelu_blockmax,
                     dim3(BATCH_N / GROUP), dim3(THREADS), 0, stream,
                     x, w, bias, out);
}